// CrossFormerAttention_53034256171298
// MI455X (gfx1250) — compile-verified
//
#include <hip/hip_runtime.h>
#include <hip/hip_bf16.h>

// ---------------------------------------------------------------------------
// CrossFormer graph attention for MI455X (gfx1250, wave32).
//
// Phase 1: Q/K/V projections via V_WMMA_F32_16X16X4_F32 (fp32 matrix cores,
//          no precision loss; phase is bandwidth-bound at ~154 MB anyway).
// Phase 2: one wave32 per edge; per-head 16-dim dot reduced with two
//          shfl_xor butterflies; scatter via global_atomic_add_f32 into an
//          L2-resident 25.6 MB output.
// ---------------------------------------------------------------------------

#define N_NODES 50000
#define N_EDGES 1600000
#define IN_DIM  128
#define HD      128   // NUM_HEADS * OUT_DIM

typedef __attribute__((ext_vector_type(2))) float v2f;
typedef __attribute__((ext_vector_type(8))) float v8f;

// ---------------------------------------------------------------------------
// Projection GEMM: C[50000 x 128] = A[50000 x 128] * B[128 x 128]
// grid.x = 3125 (M tiles of 16), grid.y = 3 (Q / K / V), block = 256 (8 waves)
// wave w computes the 16x16 tile at columns [16w, 16w+16).
// ---------------------------------------------------------------------------
__global__ void __launch_bounds__(256)
proj_wmma_f32_kernel(const float* __restrict__ h,
                     const float* __restrict__ h_add,
                     const float* __restrict__ WQ,
                     const float* __restrict__ WK,
                     const float* __restrict__ WV,
                     float* __restrict__ Qh,
                     float* __restrict__ Kh,
                     float* __restrict__ Vh) {
    const int wave  = (int)(threadIdx.x >> 5);   // 0..7  -> N tile
    const int lane  = (int)(threadIdx.x & 31);
    const int lo    = lane & 15;                 // M (A/C) or N (B/C) index
    const int hi    = lane >> 4;                 // 0 or 1: K-pair / M+8 select
    const int mbase = (int)blockIdx.x * 16;
    const int nbase = wave * 16;

    const float* __restrict__ A;
    const float* __restrict__ B;
    float*       __restrict__ C;
    if (blockIdx.y == 0)      { A = h_add; B = WQ; C = Qh; }
    else if (blockIdx.y == 1) { A = h;     B = WK; C = Kh; }
    else                      { A = h;     B = WV; C = Vh; }

    const float* __restrict__ arow = A + (size_t)(mbase + lo) * IN_DIM;
    const float* __restrict__ bcol = B + (size_t)(nbase + lo);

    v8f acc = {};
#pragma unroll 4
    for (int k = 0; k < IN_DIM; k += 4) {
        const int ka = k + 2 * hi;
        // A 16x4 fragment: lane holds A[mbase+lo][ka], A[mbase+lo][ka+1]
        const float2 af = *reinterpret_cast<const float2*>(arow + ka);
        v2f a; a.x = af.x; a.y = af.y;
        // B 4x16 fragment: lane holds B[ka][nbase+lo], B[ka+1][nbase+lo]
        v2f b;
        b.x = bcol[(size_t)ka * HD];
        b.y = bcol[(size_t)(ka + 1) * HD];
        acc = __builtin_amdgcn_wmma_f32_16x16x4_f32(
            /*neg_a=*/false, a, /*neg_b=*/false, b,
            /*c_mod=*/(short)0, acc, /*reuse_a=*/false, /*reuse_b=*/false);
    }

    // C/D layout: VGPR r -> row (r + 8*hi), col (nbase + lo)
    float* __restrict__ crow = C + (size_t)(mbase + 8 * hi) * HD + nbase + lo;
#pragma unroll
    for (int r = 0; r < 8; ++r)
        crow[(size_t)r * HD] = acc[r];
}

// ---------------------------------------------------------------------------
// Edge scatter: one wave per edge. Lane l owns dims [4l, 4l+4); lanes
// 4h..4h+3 form head h's group. score_h = sum_d K[src][h][d]*Q[dst][h][d].
// out[dst] += V[src] * score * (1/N)  via global_atomic_add_f32.
// ---------------------------------------------------------------------------
__global__ void __launch_bounds__(256)
edge_scatter_kernel(const int* __restrict__ src,
                    const int* __restrict__ dst,
                    const float* __restrict__ Qh,
                    const float* __restrict__ Kh,
                    const float* __restrict__ Vh,
                    float* __restrict__ out,
                    int n_edges, float inv_n) {
    const int gtid = (int)(blockIdx.x * blockDim.x + threadIdx.x);
    const int edge = gtid >> 5;
    const int lane = gtid & 31;
    if (edge >= n_edges) return;

    // src/dst are wave-uniform: force to SGPR so QKV bases are scalar.
    const int s = __builtin_amdgcn_readfirstlane(src[edge]);
    const int d = __builtin_amdgcn_readfirstlane(dst[edge]);

    const float4 kv = *reinterpret_cast<const float4*>(Kh + (size_t)s * HD + lane * 4);
    const float4 qv = *reinterpret_cast<const float4*>(Qh + (size_t)d * HD + lane * 4);

    float p = kv.x * qv.x + kv.y * qv.y + kv.z * qv.z + kv.w * qv.w;
    // reduce across the 4 lanes of this head's group (wave32 butterflies)
    p += __shfl_xor(p, 1, 32);
    p += __shfl_xor(p, 2, 32);
    const float score = p * inv_n;

    const float4 vv = *reinterpret_cast<const float4*>(Vh + (size_t)s * HD + lane * 4);
    float* __restrict__ op = out + (size_t)d * HD + lane * 4;
    atomicAdd(op + 0, vv.x * score);
    atomicAdd(op + 1, vv.y * score);
    atomicAdd(op + 2, vv.z * score);
    atomicAdd(op + 3, vv.w * score);
}

// ---------------------------------------------------------------------------
extern "C" void kernel_launch(void* const* d_in, const int* in_sizes, int n_in,
                              void* d_out, int out_size, void* d_ws, size_t ws_size,
                              hipStream_t stream) {
    const float* h     = (const float*)d_in[0];
    const float* h_add = (const float*)d_in[1];
    const int*   src   = (const int*)d_in[2];
    const int*   dst   = (const int*)d_in[3];
    const float* WQ    = (const float*)d_in[4];
    const float* WK    = (const float*)d_in[5];
    const float* WV    = (const float*)d_in[6];
    float* out = (float*)d_out;

    const int n_nodes = in_sizes[0] / IN_DIM;   // 50000
    const int n_edges = in_sizes[2];            // 1600000

    // Workspace: Q | K | V, each n_nodes * 128 floats (25.6 MB each).
    float* Qh = (float*)d_ws;
    float* Kh = Qh + (size_t)n_nodes * HD;
    float* Vh = Kh + (size_t)n_nodes * HD;

    // Output is accumulated with atomics: zero it every call (capture-legal).
    hipMemsetAsync(d_out, 0, (size_t)out_size * sizeof(float), stream);

    // Phase 1: projections. M tiles of 16 (50000 / 16 = 3125 exactly).
    dim3 pgrid((n_nodes + 15) / 16, 3, 1);
    proj_wmma_f32_kernel<<<pgrid, 256, 0, stream>>>(h, h_add, WQ, WK, WV, Qh, Kh, Vh);

    // Phase 2: one wave32 per edge, 8 edges per 256-thread block.
    const int waves_per_block = 256 / 32;
    const int eblocks = (n_edges + waves_per_block - 1) / waves_per_block;
    edge_scatter_kernel<<<eblocks, 256, 0, stream>>>(
        src, dst, Qh, Kh, Vh, out, n_edges, 1.0f / (float)n_nodes);
}